// TopicRouter_57690000720298
// MI455X (gfx1250) — compile-verified
//
#include <hip/hip_runtime.h>

typedef float v2f __attribute__((ext_vector_type(2)));
typedef float v8f __attribute__((ext_vector_type(8)));

#define DM   768
#define NE   8
#define NEP  16   // experts padded to 16 (WMMA N dimension)
#define TPW  16   // tokens per wave (WMMA M dimension)
#define WAVES 8
#define TPB  (WAVES * TPW)   // 128 tokens per block

__global__ __launch_bounds__(256) void
TopicRouter_router_kernel(const float* __restrict__ h,
                          const float* __restrict__ gw,
                          const float* __restrict__ gb,
                          int*   __restrict__ out_idx,
                          float* __restrict__ out_w,
                          int B)
{
    // gate_w^T, zero-padded to 16 experts, stored as interleaved (k, k+1) pairs
    // per expert: float2 element (p, e) = { gwT[2p][e], gwT[2p+1][e] }.
    __shared__ __align__(16) float lds_b[(DM / 2) * 2 * NEP];      // 48 KB
    __shared__ __align__(16) float lds_l[WAVES][TPW][NEP];         //  8 KB

    const int tid = threadIdx.x;

    // ---- Stage B (gate_w) into LDS, padded ----
    for (int i = tid; i < DM * NEP; i += 256) {
        int k = i >> 4;        // 0..767
        int e = i & 15;        // 0..15
        float v = (e < NE) ? gw[e * DM + k] : 0.0f;
        lds_b[(k >> 1) * (2 * NEP) + e * 2 + (k & 1)] = v;
    }
    __syncthreads();

    const int wave = tid >> 5;
    const int lane = tid & 31;
    const int half = lane >> 4;      // 0: K=k0,k0+1   1: K=k0+2,k0+3
    const int l15  = lane & 15;
    const int tile_base = blockIdx.x * TPB + wave * TPW;

    // A-fragment source: lane streams its own token row of h.
    const float* aptr = h + (size_t)(tile_base + l15) * DM + 2 * half;
    // B-fragment source in LDS (one b64 per step, conflict-free).
    const v2f* bptr = (const v2f*)lds_b + half * NEP + l15;

    v8f c = {0.f, 0.f, 0.f, 0.f, 0.f, 0.f, 0.f, 0.f};

#pragma unroll 8
    for (int k0 = 0; k0 < DM; k0 += 4) {
        v2f a = *(const v2f*)(aptr + k0);
        v2f b = bptr[(k0 >> 1) * NEP];
        c = __builtin_amdgcn_wmma_f32_16x16x4_f32(
                /*neg_a=*/false, a, /*neg_b=*/false, b,
                /*c_mod=*/(short)0, c, /*reuse_a=*/false, /*reuse_b=*/false);
    }

    // ---- Spill logits (C/D layout: VGPR v -> row v (lanes 0-15) / v+8 (16-31),
    //      N = lane&15) to this wave's private LDS slab ----
#pragma unroll
    for (int v = 0; v < 8; ++v) {
        lds_l[wave][v + half * 8][l15] = c[v];
    }
    __syncthreads();

    // ---- Top-2 over the 8 real experts + softmax, 1 token per lane ----
    if (lane < TPW) {
        const int token = tile_base + lane;
        float best = -INFINITY, second = -INFINITY;
        int bi = 0, si = 0;
#pragma unroll
        for (int e = 0; e < NE; ++e) {
            float v = lds_l[wave][lane][e] + gb[e];
            if (v > best)        { second = best; si = bi; best = v; bi = e; }
            else if (v > second) { second = v;    si = e; }
        }
        // softmax over [best, second]; best >= second so this is stable.
        float t  = expf(second - best);
        float r  = 1.0f / (1.0f + t);
        if (token < B) {
            out_idx[token * 2 + 0] = bi;
            out_idx[token * 2 + 1] = si;
            out_w[token * 2 + 0]   = r;
            out_w[token * 2 + 1]   = t * r;
        }
    }
}

extern "C" void kernel_launch(void* const* d_in, const int* in_sizes, int n_in,
                              void* d_out, int out_size, void* d_ws, size_t ws_size,
                              hipStream_t stream) {
    const float* h  = (const float*)d_in[0];   // [B, 768]
    const float* gw = (const float*)d_in[1];   // [8, 768]
    const float* gb = (const float*)d_in[2];   // [8]
    const int B = in_sizes[0] / DM;            // 131072

    // d_out = concat(topk_idx [B,2] int32, weights [B,2] float32)
    int*   out_idx = (int*)d_out;
    float* out_w   = (float*)d_out + (size_t)B * 2;

    const int blocks = B / TPB;                // 131072 / 128 = 1024 (exact)
    TopicRouter_router_kernel<<<blocks, 256, 0, stream>>>(h, gw, gb, out_idx, out_w, B);
}